// BinsChamferLoss_17712445128930
// MI455X (gfx1250) — compile-verified
//
#include <hip/hip_runtime.h>
#include <stdint.h>

typedef __attribute__((ext_vector_type(2))) float v2f;
typedef __attribute__((ext_vector_type(8))) float v8f;

#define N_IMG 16
#define D_BINS 128
#define HW 49152            // 192*256
#define PB1 24              // blocks per image, kernel 1 (2048 px/block)
#define PIX_PER_BLK 2048
#define S2 16               // pixel slices per image, kernel 2 (3072 px/slice)

// ws layout (floats):
//   [0,384)         loss1 partials   [img][PB1]
//   [384,768)       count partials   [img][PB1]
//   [768,768+32768) binmin partials  [img][slice][128]
#define WS_L1   0
#define WS_CNT  384
#define WS_BMIN 768

#define FBIG 3.402823466e38f

// Two independent branchless lower-bound searches, interleaved so the
// dependent ds_load chains pipeline (2 loads in flight per wait).
// sbins has a +INF sentinel at index 128 -> upper-neighbor read is always safe.
__device__ __forceinline__ void nearest2(const float* __restrict__ sb,
                                         float g0, float g1,
                                         float& r0, float& r1) {
    int lo0 = 0, lo1 = 0;
    #pragma unroll
    for (int s = 64; s > 0; s >>= 1) {
        float a0 = sb[lo0 + s];
        float a1 = sb[lo1 + s];
        lo0 += (a0 <= g0) ? s : 0;
        lo1 += (a1 <= g1) ? s : 0;
    }
    float lw0 = sb[lo0], hi0 = sb[lo0 + 1];
    float lw1 = sb[lo1], hi1 = sb[lo1 + 1];
    r0 = fminf(g0 - lw0, hi0 - g0);
    r1 = fminf(g1 - lw1, hi1 - g1);
}

// Kernel 1: per-pixel nearest-bin distance (loss1 partial) + mask count.
// grid (PB1, N_IMG), 256 threads. WMMA f32 16x16x4 used as exact fp32 accumulator.
__global__ void chamfer_k1(const float* __restrict__ bins,
                           const float* __restrict__ tgt,
                           const unsigned char* __restrict__ msk,
                           float* __restrict__ ws) {
    __shared__ float sbins[D_BINS + 4];   // +sentinel
    __shared__ float sred[256];

    const int tid  = threadIdx.x;
    const int lane = tid & 31;
    const int wave = tid >> 5;
    const int blk  = blockIdx.x;
    const int img  = blockIdx.y;

    // --- async-stage the 128 bins of this image into LDS (wave 0, 16 B/lane) ---
    if (wave == 0) {
        const float* gsrc = bins + (size_t)img * D_BINS + lane * 4;
        unsigned ldsOff = (unsigned)(uintptr_t)(&sbins[lane * 4]);
        asm volatile("global_load_async_to_lds_b128 %0, %1, off"
                     :: "v"(ldsOff), "v"(gsrc) : "memory");
    }
    if (tid == 0) sbins[D_BINS] = FBIG;   // sentinel: makes search fully branchless
    asm volatile("s_wait_asynccnt 0" ::: "memory");
    __syncthreads();

    const float b0 = sbins[0];
    const size_t imgBase = (size_t)img * HW + (size_t)blk * PIX_PER_BLK;

    v2f ones; ones[0] = 1.0f; ones[1] = 1.0f;
    v8f acc = {};
    int cnt = 0;

    #pragma unroll
    for (int it = 0; it < PIX_PER_BLK / 512; ++it) {   // 4 iters, 2 px/thread, coalesced b64
        const size_t p = imgBase + (size_t)it * 512 + (size_t)tid * 2;
        float t0 = tgt[p], t1 = tgt[p + 1];
        unsigned char m0 = msk[p], m1 = msk[p + 1];
        cnt += (int)(m0 != 0) + (int)(m1 != 0);
        float g0 = fmaxf(m0 ? t0 : 0.0f, b0);
        float g1 = fmaxf(m1 ? t1 : 0.0f, b0);
        v2f bb;
        float d0, d1;
        nearest2(sbins, g0, g1, d0, d1);
        bb[0] = d0; bb[1] = d1;
        // D = ones(16x4) x B(4x16) + D : every row of D becomes the running
        // column sums; sum(row 0) == exact fp32 sum of all values ever fed in B.
        acc = __builtin_amdgcn_wmma_f32_16x16x4_f32(false, ones, false, bb,
                                                    (short)0, acc, false, false);
    }

    // extract: row 0 of D lives in acc[0] on lanes 0..15 of each wave
    if (lane < 16) sred[wave * 16 + lane] = acc[0];
    __syncthreads();
    if (tid == 0) {
        float s = 0.0f;
        for (int k = 0; k < 128; ++k) s += sred[k];
        ws[WS_L1 + img * PB1 + blk] = s;
    }
    __syncthreads();
    sred[tid] = (float)cnt;
    __syncthreads();
    if (tid == 0) {
        float c = 0.0f;
        for (int k = 0; k < 256; ++k) c += sred[k];
        ws[WS_CNT + img * PB1 + blk] = c;
    }
}

// Kernel 2: per-bin min over a pixel slice. grid (S2, N_IMG), 256 threads.
// Stage 512 g-values to LDS (coalesced), 2 threads per bin scan with broadcast reads.
__global__ void chamfer_k2(const float* __restrict__ bins,
                           const float* __restrict__ tgt,
                           const unsigned char* __restrict__ msk,
                           float* __restrict__ ws) {
    __shared__ float gch[512];
    __shared__ float sred[256];

    const int tid = threadIdx.x;
    const int sl  = blockIdx.x;
    const int img = blockIdx.y;
    const int bin = tid & (D_BINS - 1);
    const int par = tid >> 7;                     // which half of each chunk we scan

    const float b0 = bins[(size_t)img * D_BINS];
    const float bv = bins[(size_t)img * D_BINS + bin];
    const size_t sliceBase = (size_t)img * HW + (size_t)sl * (HW / S2);

    float runmin = FBIG;
    for (int ch = 0; ch < (HW / S2) / 512; ++ch) {     // 6 chunks of 512 px
        const size_t p = sliceBase + (size_t)ch * 512 + (size_t)tid * 2;
        float t0 = tgt[p], t1 = tgt[p + 1];
        unsigned char m0 = msk[p], m1 = msk[p + 1];
        float g0 = fmaxf(m0 ? t0 : 0.0f, b0);
        float g1 = fmaxf(m1 ? t1 : 0.0f, b0);
        __syncthreads();                               // previous chunk fully consumed
        gch[tid * 2]     = g0;
        gch[tid * 2 + 1] = g1;
        __syncthreads();
        #pragma unroll 8
        for (int q = par; q < 512; q += 2)             // same q across a wave -> LDS broadcast
            runmin = fminf(runmin, fabsf(gch[q] - bv));
    }

    sred[tid] = runmin;
    __syncthreads();
    if (tid < D_BINS)
        ws[WS_BMIN + ((size_t)img * S2 + sl) * D_BINS + tid] =
            fminf(sred[tid], sred[tid + D_BINS]);
}

// Kernel 3: final deterministic reduction. grid N_IMG, 128 threads.
__global__ void chamfer_k3(const float* __restrict__ ws, float* __restrict__ out) {
    __shared__ float sred[D_BINS];
    const int img = blockIdx.x;
    const int tid = threadIdx.x;

    float m = FBIG;
    for (int s = 0; s < S2; ++s)
        m = fminf(m, ws[WS_BMIN + ((size_t)img * S2 + s) * D_BINS + tid]);
    sred[tid] = m;
    __syncthreads();
    if (tid == 0) {
        float loss2 = 0.0f;
        for (int k = 0; k < D_BINS; ++k) loss2 += sred[k];
        float loss1 = 0.0f;
        for (int b = 0; b < PB1; ++b) loss1 += ws[WS_L1 + img * PB1 + b];
        float cnt = 0.0f;
        for (int k = 0; k < N_IMG * PB1; ++k) cnt += ws[WS_CNT + k];
        out[img] = (loss1 + loss2) / cnt;
    }
}

extern "C" void kernel_launch(void* const* d_in, const int* in_sizes, int n_in,
                              void* d_out, int out_size, void* d_ws, size_t ws_size,
                              hipStream_t stream) {
    (void)in_sizes; (void)n_in; (void)out_size; (void)ws_size;
    const float*         bins = (const float*)d_in[0];          // (16,128) fp32, sorted
    const float*         tgt  = (const float*)d_in[1];          // (16,1,192,256) fp32
    const unsigned char* msk  = (const unsigned char*)d_in[2];  // (16,1,192,256) bool (1 B)
    float* ws  = (float*)d_ws;
    float* out = (float*)d_out;

    chamfer_k1<<<dim3(PB1, N_IMG), 256, 0, stream>>>(bins, tgt, msk, ws);
    chamfer_k2<<<dim3(S2,  N_IMG), 256, 0, stream>>>(bins, tgt, msk, ws);
    chamfer_k3<<<N_IMG, D_BINS, 0, stream>>>(ws, out);
}